// ConvLRULayer_29368986370552
// MI455X (gfx1250) — compile-verified
//
#include <hip/hip_runtime.h>
#include <hip/hip_bf16.h>
#include <math.h>

// ---------------------------------------------------------------------------
// ConvLRU layer for MI455X (gfx1250, wave32, WMMA).
// All GEMM-like work (DFT2, channel mixes) runs on V_WMMA_F32_16X16X4_F32.
// ---------------------------------------------------------------------------

typedef __attribute__((ext_vector_type(2))) float v2f;
typedef __attribute__((ext_vector_type(8))) float v8f;

#define BB 4
#define LL 64
#define CC 64
#define HH 32
#define WWW 32
#define HW 1024            // H*W
#define CHW 65536          // C*HW
#define PLANE 16777216     // B*L*C*HW

// workspace layout (float offsets)
#define WS_LAMR 0
#define WS_LAMI 2048
#define WS_GAM  4096
#define WS_FRE  6144
#define WS_FIM  7168
#define WS_IRE  8192
#define WS_IIM  9216
#define WS_HRE  16384
#define WS_HIM  (WS_HRE + PLANE)
#define WS_GRE  (WS_HIM + PLANE)
#define WS_GIM  (WS_GRE + PLANE)

// D = A(16x4 f32) * B(4x16 f32) + C(16x16 f32), wave32 WMMA.
__device__ __forceinline__ v8f wmma4(v2f a, v2f b, v8f c) {
  return __builtin_amdgcn_wmma_f32_16x16x4_f32(false, a, false, b, (short)0, c,
                                               false, false);
}

// ---------------------------------------------------------------------------
// Kernel 1: lambda / gamma / DFT matrices.
// params_log layout: rows 0..63 = nu_log, 64..127 = theta_log, 128..191 = gamma_log
// each row has H=32 entries.
// ---------------------------------------------------------------------------
__global__ __launch_bounds__(256) void kprep(const float* __restrict__ plog,
                                             float* __restrict__ ws) {
  const int tid = threadIdx.x;
  for (int i = tid; i < 2048; i += 256) {
    const int c = i >> 5, hr = i & 31;
    const float nu = expf(plog[c * 32 + hr]);
    const float th = expf(plog[(64 + c) * 32 + hr]);
    const float gm = expf(plog[(128 + c) * 32 + hr]);
    const float mag = expf(-nu);
    ws[WS_LAMR + i] = mag * cosf(th);
    ws[WS_LAMI + i] = mag * sinf(th);
    ws[WS_GAM + i] = gm;
  }
  for (int i = tid; i < 1024; i += 256) {
    const int j = i >> 5, k = i & 31;
    const float ang = -6.283185307179586f * (float)((j * k) & 31) * (1.0f / 32.0f);
    ws[WS_FRE + i] = cosf(ang);
    ws[WS_FIM + i] = sinf(ang);
    ws[WS_IRE + i] = cosf(ang) * (1.0f / 32.0f);   // cos even
    ws[WS_IIM + i] = -sinf(ang) * (1.0f / 32.0f);  // conj, scaled 1/N per axis
  }
}

// ---------------------------------------------------------------------------
// Kernel 2: forward DFT2 per (b,l,c) tile: H = F * X * F  (F symmetric 32x32).
// 128 threads = 4 waves, each wave owns one 16x16 quadrant.
// ---------------------------------------------------------------------------
__global__ __launch_bounds__(128) void kfwd_dft(const float* __restrict__ x,
                                                const float* __restrict__ ws,
                                                float* __restrict__ hre,
                                                float* __restrict__ him) {
  __shared__ float Xs[1024];
  __shared__ float Tre[1024];
  __shared__ float Tim[1024];
  const float* Fre = ws + WS_FRE;
  const float* Fim = ws + WS_FIM;
  const int tid = threadIdx.x;
  const int tile = blockIdx.x;
  const float* xt = x + (size_t)tile * HW;
  for (int i = tid; i < 1024; i += 128) Xs[i] = xt[i];
  __syncthreads();

  const int w = tid >> 5;
  const int lane = tid & 31;
  const int hi = lane >> 4, lo = lane & 15;
  const int mbase = (w >> 1) * 16, nbase = (w & 1) * 16;

  // Stage A: T = X * F  (X real)
  v8f tR = {0, 0, 0, 0, 0, 0, 0, 0};
  v8f tI = tR;
  for (int k0 = 0; k0 < 32; k0 += 4) {
    const int kr = k0 + 2 * hi;
    v2f a;  a.x = Xs[(mbase + lo) * 32 + kr];        a.y = Xs[(mbase + lo) * 32 + kr + 1];
    v2f bR; bR.x = Fre[kr * 32 + nbase + lo];        bR.y = Fre[(kr + 1) * 32 + nbase + lo];
    v2f bI; bI.x = Fim[kr * 32 + nbase + lo];        bI.y = Fim[(kr + 1) * 32 + nbase + lo];
    tR = wmma4(a, bR, tR);
    tI = wmma4(a, bI, tI);
  }
  {
    const int m0 = mbase + 8 * hi, n = nbase + lo;
#pragma unroll
    for (int r = 0; r < 8; ++r) {
      Tre[(m0 + r) * 32 + n] = tR[r];
      Tim[(m0 + r) * 32 + n] = tI[r];
    }
  }
  __syncthreads();

  // Stage B: H = F * T  (complex)
  v8f hR = {0, 0, 0, 0, 0, 0, 0, 0};
  v8f hI = hR;
  for (int k0 = 0; k0 < 32; k0 += 4) {
    const int kr = k0 + 2 * hi;
    v2f aR; aR.x = Fre[(mbase + lo) * 32 + kr];  aR.y = Fre[(mbase + lo) * 32 + kr + 1];
    v2f aI; aI.x = Fim[(mbase + lo) * 32 + kr];  aI.y = Fim[(mbase + lo) * 32 + kr + 1];
    v2f aIn; aIn.x = -aI.x; aIn.y = -aI.y;       // f32 WMMA has no A-neg modifier
    v2f bR; bR.x = Tre[kr * 32 + nbase + lo];    bR.y = Tre[(kr + 1) * 32 + nbase + lo];
    v2f bI; bI.x = Tim[kr * 32 + nbase + lo];    bI.y = Tim[(kr + 1) * 32 + nbase + lo];
    hR = wmma4(aR, bR, hR);  hR = wmma4(aIn, bI, hR);
    hI = wmma4(aR, bI, hI);  hI = wmma4(aI, bR, hI);
  }
  {
    const size_t base = (size_t)tile * HW;
    const int m0 = mbase + 8 * hi, n = nbase + lo;
#pragma unroll
    for (int r = 0; r < 8; ++r) {
      hre[base + (m0 + r) * 32 + n] = hR[r];
      him[base + (m0 + r) * 32 + n] = hI[r];
    }
  }
}

// ---------------------------------------------------------------------------
// Kernel 3: channel mix with Wb (complex), + bb, * gamma.
// Per (b,l): Out(C x HW) = Wb(C x C) * H(C x HW).  8 blocks, 128-col slab each; 16 cols per wave.
// ---------------------------------------------------------------------------
__global__ __launch_bounds__(256) void kmix_b(const float* __restrict__ hre,
                                              const float* __restrict__ him,
                                              const float* __restrict__ Wbr,
                                              const float* __restrict__ Wbi,
                                              const float* __restrict__ bbr,
                                              const float* __restrict__ bbi,
                                              const float* __restrict__ ws,
                                              float* __restrict__ gre,
                                              float* __restrict__ gim) {
  __shared__ float Wr[4096];
  __shared__ float Wi[4096];
  const int tid = threadIdx.x;
  for (int i = tid; i < 4096; i += 256) { Wr[i] = Wbr[i]; Wi[i] = Wbi[i]; }
  __syncthreads();

  const int bl = blockIdx.x >> 3;
  const int slab = blockIdx.x & 7;
  const int w = tid >> 5, lane = tid & 31, hi = lane >> 4, lo = lane & 15;
  const int nbase = slab * 128 + w * 16;
  const float* Hr = hre + (size_t)bl * CHW;
  const float* Hi = him + (size_t)bl * CHW;
  const float* gamma = ws + WS_GAM;

  const v8f zero = {0, 0, 0, 0, 0, 0, 0, 0};
  v8f accR[4] = {zero, zero, zero, zero};
  v8f accI[4] = {zero, zero, zero, zero};

  for (int k0 = 0; k0 < 64; k0 += 4) {
    const int kr = k0 + 2 * hi;
    v2f bR; bR.x = Hr[kr * HW + nbase + lo];  bR.y = Hr[(kr + 1) * HW + nbase + lo];
    v2f bI; bI.x = Hi[kr * HW + nbase + lo];  bI.y = Hi[(kr + 1) * HW + nbase + lo];
#pragma unroll
    for (int mt = 0; mt < 4; ++mt) {
      const int m = mt * 16 + lo;
      v2f ar; ar.x = Wr[m * 64 + kr];  ar.y = Wr[m * 64 + kr + 1];
      v2f ai; ai.x = Wi[m * 64 + kr];  ai.y = Wi[m * 64 + kr + 1];
      v2f an; an.x = -ai.x; an.y = -ai.y;
      accR[mt] = wmma4(ar, bR, accR[mt]);
      accR[mt] = wmma4(an, bI, accR[mt]);
      accI[mt] = wmma4(ar, bI, accI[mt]);
      accI[mt] = wmma4(ai, bR, accI[mt]);
    }
  }
#pragma unroll
  for (int mt = 0; mt < 4; ++mt) {
    const int m0 = mt * 16 + 8 * hi;
    const int n = nbase + lo;
    const int hrow = n >> 5;
#pragma unroll
    for (int r = 0; r < 8; ++r) {
      const int o = m0 + r;
      const float g = gamma[o * 32 + hrow];
      gre[(size_t)bl * CHW + o * HW + n] = (accR[mt][r] + bbr[o]) * g;
      gim[(size_t)bl * CHW + o * HW + n] = (accI[mt][r] + bbi[o]) * g;
    }
  }
}

// ---------------------------------------------------------------------------
// Kernel 4: linear recurrence over L (in place on g), emits last hidden state.
// One thread per (b,c,hw) sequence; fully coalesced across hw.
// ---------------------------------------------------------------------------
__global__ __launch_bounds__(256) void kscan(float* __restrict__ gre,
                                             float* __restrict__ gim,
                                             const float* __restrict__ ws,
                                             float* __restrict__ out_last) {
  const int idx = blockIdx.x * 256 + threadIdx.x;  // 0..B*C*HW-1
  const int b = idx >> 16;
  const int rem = idx & 65535;
  const int c = rem >> 10;
  const int hw = rem & 1023;
  const int hrow = hw >> 5;
  const float lr = ws[WS_LAMR + c * 32 + hrow];
  const float li = ws[WS_LAMI + c * 32 + hrow];
  size_t ptr = ((size_t)(b * LL) * CC + c) * HW + hw;
  float pr = 0.f, pi = 0.f;
  for (int l = 0; l < LL; ++l) {
    if (l + 1 < LL) {  // uniform branch; prefetch next timestep's lines
      __builtin_prefetch(gre + ptr + CHW, 0, 0);
      __builtin_prefetch(gim + ptr + CHW, 0, 0);
    }
    const float hr = gre[ptr];
    const float hi = gim[ptr];
    const float nr = fmaf(lr, pr, fmaf(-li, pi, hr));
    const float ni = fmaf(lr, pi, fmaf(li, pr, hi));
    gre[ptr] = nr;
    gim[ptr] = ni;
    pr = nr;
    pi = ni;
    ptr += CHW;
  }
  // last_hidden_out: complex64 (B,1,C,H,W), interleaved re/im
  const size_t o = ((size_t)(b * CC + c) * HW + hw) * 2;
  out_last[o] = pr;
  out_last[o + 1] = pi;
}

// ---------------------------------------------------------------------------
// Kernel 5: inverse DFT2 per (b,l,c) tile: out = Fi * Hc * Fi (Fi = conj(F)/32).
// ---------------------------------------------------------------------------
__global__ __launch_bounds__(128) void kinv_dft(const float* __restrict__ gre,
                                                const float* __restrict__ gim,
                                                const float* __restrict__ ws,
                                                float* __restrict__ hre,
                                                float* __restrict__ him) {
  __shared__ float Hre[1024];
  __shared__ float Him[1024];
  __shared__ float Tre[1024];
  __shared__ float Tim[1024];
  const float* FiR = ws + WS_IRE;
  const float* FiI = ws + WS_IIM;
  const int tid = threadIdx.x;
  const int tile = blockIdx.x;
  const size_t base = (size_t)tile * HW;
  for (int i = tid; i < 1024; i += 128) {
    Hre[i] = gre[base + i];
    Him[i] = gim[base + i];
  }
  __syncthreads();

  const int w = tid >> 5;
  const int lane = tid & 31;
  const int hi = lane >> 4, lo = lane & 15;
  const int mbase = (w >> 1) * 16, nbase = (w & 1) * 16;

  // Stage A: T = Hc * Fi  (complex x complex)
  v8f tR = {0, 0, 0, 0, 0, 0, 0, 0};
  v8f tI = tR;
  for (int k0 = 0; k0 < 32; k0 += 4) {
    const int kr = k0 + 2 * hi;
    v2f aR; aR.x = Hre[(mbase + lo) * 32 + kr];  aR.y = Hre[(mbase + lo) * 32 + kr + 1];
    v2f aI; aI.x = Him[(mbase + lo) * 32 + kr];  aI.y = Him[(mbase + lo) * 32 + kr + 1];
    v2f an; an.x = -aI.x; an.y = -aI.y;
    v2f bR; bR.x = FiR[kr * 32 + nbase + lo];    bR.y = FiR[(kr + 1) * 32 + nbase + lo];
    v2f bI; bI.x = FiI[kr * 32 + nbase + lo];    bI.y = FiI[(kr + 1) * 32 + nbase + lo];
    tR = wmma4(aR, bR, tR);  tR = wmma4(an, bI, tR);
    tI = wmma4(aR, bI, tI);  tI = wmma4(aI, bR, tI);
  }
  {
    const int m0 = mbase + 8 * hi, n = nbase + lo;
#pragma unroll
    for (int r = 0; r < 8; ++r) {
      Tre[(m0 + r) * 32 + n] = tR[r];
      Tim[(m0 + r) * 32 + n] = tI[r];
    }
  }
  __syncthreads();

  // Stage B: out = Fi * T (complex)
  v8f hR = {0, 0, 0, 0, 0, 0, 0, 0};
  v8f hI = hR;
  for (int k0 = 0; k0 < 32; k0 += 4) {
    const int kr = k0 + 2 * hi;
    v2f aR; aR.x = FiR[(mbase + lo) * 32 + kr];  aR.y = FiR[(mbase + lo) * 32 + kr + 1];
    v2f aI; aI.x = FiI[(mbase + lo) * 32 + kr];  aI.y = FiI[(mbase + lo) * 32 + kr + 1];
    v2f an; an.x = -aI.x; an.y = -aI.y;
    v2f bR; bR.x = Tre[kr * 32 + nbase + lo];    bR.y = Tre[(kr + 1) * 32 + nbase + lo];
    v2f bI; bI.x = Tim[kr * 32 + nbase + lo];    bI.y = Tim[(kr + 1) * 32 + nbase + lo];
    hR = wmma4(aR, bR, hR);  hR = wmma4(an, bI, hR);
    hI = wmma4(aR, bI, hI);  hI = wmma4(aI, bR, hI);
  }
  {
    const int m0 = mbase + 8 * hi, n = nbase + lo;
#pragma unroll
    for (int r = 0; r < 8; ++r) {
      hre[base + (m0 + r) * 32 + n] = hR[r];
      him[base + (m0 + r) * 32 + n] = hI[r];
    }
  }
}

// ---------------------------------------------------------------------------
// Kernel 6: channel mix with Wc, real part only:  y = Wc_r*Hre - Wc_i*Him + bc_r.
// ---------------------------------------------------------------------------
__global__ __launch_bounds__(256) void kmix_c(const float* __restrict__ hre,
                                              const float* __restrict__ him,
                                              const float* __restrict__ Wcr,
                                              const float* __restrict__ Wci,
                                              const float* __restrict__ bcr,
                                              float* __restrict__ y) {
  __shared__ float Wr[4096];
  __shared__ float Wi[4096];
  const int tid = threadIdx.x;
  for (int i = tid; i < 4096; i += 256) { Wr[i] = Wcr[i]; Wi[i] = Wci[i]; }
  __syncthreads();

  const int bl = blockIdx.x >> 3;
  const int slab = blockIdx.x & 7;
  const int w = tid >> 5, lane = tid & 31, hi = lane >> 4, lo = lane & 15;
  const int nbase = slab * 128 + w * 16;
  const float* Hr = hre + (size_t)bl * CHW;
  const float* Hi = him + (size_t)bl * CHW;

  const v8f zero = {0, 0, 0, 0, 0, 0, 0, 0};
  v8f accR[4] = {zero, zero, zero, zero};

  for (int k0 = 0; k0 < 64; k0 += 4) {
    const int kr = k0 + 2 * hi;
    v2f bR; bR.x = Hr[kr * HW + nbase + lo];  bR.y = Hr[(kr + 1) * HW + nbase + lo];
    v2f bI; bI.x = Hi[kr * HW + nbase + lo];  bI.y = Hi[(kr + 1) * HW + nbase + lo];
#pragma unroll
    for (int mt = 0; mt < 4; ++mt) {
      const int m = mt * 16 + lo;
      v2f ar; ar.x = Wr[m * 64 + kr];  ar.y = Wr[m * 64 + kr + 1];
      v2f an; an.x = -Wi[m * 64 + kr]; an.y = -Wi[m * 64 + kr + 1];
      accR[mt] = wmma4(ar, bR, accR[mt]);
      accR[mt] = wmma4(an, bI, accR[mt]);
    }
  }
#pragma unroll
  for (int mt = 0; mt < 4; ++mt) {
    const int m0 = mt * 16 + 8 * hi;
    const int n = nbase + lo;
#pragma unroll
    for (int r = 0; r < 8; ++r) {
      const int o = m0 + r;
      y[(size_t)bl * CHW + o * HW + n] = accR[mt][r] + bcr[o];
    }
  }
}

// ---------------------------------------------------------------------------
// Kernel 7: LayerNorm over (H,W) + affine + residual.  One block per (b,l,c).
// ---------------------------------------------------------------------------
__global__ __launch_bounds__(256) void kln(const float* __restrict__ y,
                                           const float* __restrict__ x,
                                           const float* __restrict__ lnw,
                                           const float* __restrict__ lnb,
                                           float* __restrict__ out) {
  __shared__ float red1[8];
  __shared__ float red2[8];
  const size_t base = (size_t)blockIdx.x * HW;
  const int tid = threadIdx.x;
  float v[4];
  float s = 0.f, s2 = 0.f;
#pragma unroll
  for (int j = 0; j < 4; ++j) {
    v[j] = y[base + tid + j * 256];
    s += v[j];
    s2 += v[j] * v[j];
  }
#pragma unroll
  for (int o = 16; o > 0; o >>= 1) {  // wave32 reduction
    s += __shfl_down(s, o, 32);
    s2 += __shfl_down(s2, o, 32);
  }
  const int wv = tid >> 5, lane = tid & 31;
  if (lane == 0) { red1[wv] = s; red2[wv] = s2; }
  __syncthreads();
  if (tid == 0) {
    float a = 0.f, b = 0.f;
#pragma unroll
    for (int i = 0; i < 8; ++i) { a += red1[i]; b += red2[i]; }
    const float mu = a * (1.0f / 1024.0f);
    const float var = b * (1.0f / 1024.0f) - mu * mu;
    red1[0] = mu;
    red2[0] = rsqrtf(var + 1e-5f);
  }
  __syncthreads();
  const float mu = red1[0];
  const float rs = red2[0];
#pragma unroll
  for (int j = 0; j < 4; ++j) {
    const int i = tid + j * 256;
    out[base + i] = (v[j] - mu) * rs * lnw[i] + lnb[i] + x[base + i];
  }
}

// ---------------------------------------------------------------------------
extern "C" void kernel_launch(void* const* d_in, const int* in_sizes, int n_in,
                              void* d_out, int out_size, void* d_ws, size_t ws_size,
                              hipStream_t stream) {
  const float* x    = (const float*)d_in[0];
  const float* plog = (const float*)d_in[1];
  const float* Wbr  = (const float*)d_in[2];
  const float* Wbi  = (const float*)d_in[3];
  const float* bbr  = (const float*)d_in[4];
  const float* bbi  = (const float*)d_in[5];
  const float* Wcr  = (const float*)d_in[6];
  const float* Wci  = (const float*)d_in[7];
  const float* bcr  = (const float*)d_in[8];
  // d_in[9] = bc_i is dropped by the real() in the reference.
  const float* lnw  = (const float*)d_in[10];
  const float* lnb  = (const float*)d_in[11];

  float* out = (float*)d_out;
  float* out_last = out + PLANE;  // complex64 (B,1,C,H,W) interleaved

  float* ws  = (float*)d_ws;
  float* hre = ws + WS_HRE;
  float* him = ws + WS_HIM;
  float* gre = ws + WS_GRE;
  float* gim = ws + WS_GIM;

  kprep<<<1, 256, 0, stream>>>(plog, ws);
  kfwd_dft<<<BB * LL * CC, 128, 0, stream>>>(x, ws, hre, him);
  kmix_b<<<BB * LL * 8, 256, 0, stream>>>(hre, him, Wbr, Wbi, bbr, bbi, ws, gre, gim);
  kscan<<<(BB * CC * HW) / 256, 256, 0, stream>>>(gre, gim, ws, out_last);
  kinv_dft<<<BB * LL * CC, 128, 0, stream>>>(gre, gim, ws, hre, him);
  kmix_c<<<BB * LL * 8, 256, 0, stream>>>(hre, him, Wcr, Wci, bcr, gre);
  kln<<<BB * LL * CC, 256, 0, stream>>>(gre, x, lnw, lnb, out);
}